// DynamicConv_21552145891626
// MI455X (gfx1250) — compile-verified
//
#include <hip/hip_runtime.h>

typedef float v2f __attribute__((ext_vector_type(2)));
typedef float v8f __attribute__((ext_vector_type(8)));

#define IN_C 32
#define CH 35
#define HID 8
#define PCOUNT 369
#define OFF_W1 0
#define OFF_B1 280
#define OFF_W2 288
#define OFF_B2 352
#define OFF_W3 360
#define OFF_B3 368
#define LROW 18   // padded LDS row stride in floats (conflict-free, even)

// Single-instruction ReLU: v_med3_f32(x, 0, +inf)
__device__ __forceinline__ float relu1(float x) {
  return __builtin_amdgcn_fmed3f(x, 0.0f, __builtin_inff());
}

// ---------------------------------------------------------------------------
// Kernel 1: params[p, :] = prop_feats[p, :] @ W_dyn + b_dyn   (128 x 369)
// Tiny (3 MFLOP) -- plain VALU, coalesced on W_dyn columns.
// ---------------------------------------------------------------------------
__global__ __launch_bounds__(256) void dynconv_params(
    const float* __restrict__ prop, const float* __restrict__ W,
    const float* __restrict__ bias, float* __restrict__ params, int numProp) {
  int idx = blockIdx.x * blockDim.x + threadIdx.x;
  int total = numProp * PCOUNT;
  if (idx >= total) return;
  int p = idx / PCOUNT;
  int c = idx - p * PCOUNT;
  float acc = bias[c];
  const float* pr = prop + p * IN_C;
#pragma unroll
  for (int k = 0; k < IN_C; ++k)
    acc = fmaf(pr[k], W[k * PCOUNT + c], acc);
  params[idx] = acc;
}

// ---------------------------------------------------------------------------
// Kernel 2: per proposal-pair MLP over all feature rows, WMMA f32 16x16x4.
// One wave = one 16-row tile x one proposal pair per iteration.
// Tile columns 0-7 -> hidden units of p0, columns 8-15 -> p1.
// ---------------------------------------------------------------------------
__global__ __launch_bounds__(256) void dynconv_mlp(
    const float* __restrict__ feats, const float* __restrict__ params,
    float* __restrict__ out, int numAll, int numTiles) {
  __shared__ float smem[8 * 16 * LROW];
  const int lane = threadIdx.x & 31;
  const int wave = threadIdx.x >> 5;
  float* lds = smem + wave * 16 * LROW;

  const int p0 = blockIdx.x * 2;
  const int p1 = p0 + 1;

  const int n   = lane & 15;      // tile column this lane owns in B/C/D layout
  const int hi  = lane >> 4;      // upper lane half -> upper K pair in A/B layout
  const int hN  = n & 7;          // hidden-unit index within its proposal
  const bool loCol = (n < 8);
  const float* prmN = params + (size_t)(loCol ? p0 : p1) * PCOUNT;

  // ---- Layer-1 B matrix: w1 (35x8 per proposal), 9 K-chunks of 4, zero-pad K=35.
  v2f B1[9];
#pragma unroll
  for (int kc = 0; kc < 9; ++kc) {
    int c0 = 4 * kc + 2 * hi;
    int c1 = c0 + 1;
    float x = prmN[OFF_W1 + ((c0 < CH) ? c0 : 0) * HID + hN];
    float y = prmN[OFF_W1 + ((c1 < CH) ? c1 : 0) * HID + hN];
    B1[kc].x = (c0 < CH) ? x : 0.f;
    B1[kc].y = (c1 < CH) ? y : 0.f;
  }

  // ---- Layer-2 B matrix: block-diagonal 16x16 (w2[p0] TL, w2[p1] BR), 4 K-chunks.
  v2f B2[4];
#pragma unroll
  for (int kc = 0; kc < 4; ++kc) {
    int k0 = 4 * kc + 2 * hi;
#pragma unroll
    for (int j = 0; j < 2; ++j) {
      int k = k0 + j;
      float v = prmN[OFF_W2 + (k & 7) * HID + hN];
      v = ((k < 8) == loCol) ? v : 0.f;
      if (j == 0) B2[kc].x = v; else B2[kc].y = v;
    }
  }

  const float bias1 = prmN[OFF_B1 + hN];
  const float bias2 = prmN[OFF_B2 + hN];

  // ---- Layer-3 per-lane weights: lane half selects proposal.
  const float* prmL = params + (size_t)(hi ? p1 : p0) * PCOUNT;
  float w3r[8];
#pragma unroll
  for (int h = 0; h < 8; ++h) w3r[h] = prmL[OFF_W3 + h];
  const float b3 = prmL[OFF_B3];

  const int row = lane & 15;      // tile row this lane owns in A layout
  const int waveStride = gridDim.y * 8;

  for (int tile = blockIdx.y * 8 + wave; tile < numTiles; tile += waveStride) {
    const int aBase = tile * 16;
    int a = aBase + row;
    if (a > numAll - 1) a = numAll - 1;          // tail clamp (branch-free)
    const float* frow = feats + (size_t)a * CH;

    // ---- Layer 1: D = relu(feats_tile @ w1_pair + b1)
    v8f d;
#pragma unroll
    for (int r = 0; r < 8; ++r) d[r] = bias1;
#pragma unroll
    for (int kc = 0; kc < 9; ++kc) {
      int c0 = 4 * kc + 2 * hi;
      v2f A;
      A.x = frow[c0];
      if (kc < 8) {
        A.y = frow[c0 + 1];
      } else {
        float t = frow[hi ? c0 : (c0 + 1)];      // never reads past column 34
        A.y = hi ? 0.f : t;                      // K=35 zero pad
      }
      d = __builtin_amdgcn_wmma_f32_16x16x4_f32(false, A, false, B1[kc],
                                                (short)0, d, false, false);
    }
#pragma unroll
    for (int r = 0; r < 8; ++r) d[r] = relu1(d[r]);

    // ---- D-layout -> A-layout relayout through wave-private LDS.
#pragma unroll
    for (int r = 0; r < 8; ++r) lds[(r + 8 * hi) * LROW + n] = d[r];

    // ---- Layer 2: E = relu(f1 @ blockdiag(w2) + b2)
    v8f e;
#pragma unroll
    for (int r = 0; r < 8; ++r) e[r] = bias2;
#pragma unroll
    for (int kc = 0; kc < 4; ++kc) {
      int k0 = 4 * kc + 2 * hi;
      v2f A2;
      A2.x = lds[row * LROW + k0];
      A2.y = lds[row * LROW + k0 + 1];
      e = __builtin_amdgcn_wmma_f32_16x16x4_f32(false, A2, false, B2[kc],
                                                (short)0, e, false, false);
    }
#pragma unroll
    for (int r = 0; r < 8; ++r) e[r] = relu1(e[r]);
#pragma unroll
    for (int r = 0; r < 8; ++r) lds[(r + 8 * hi) * LROW + n] = e[r];

    // ---- Layer 3: per-lane 8-FMA dot; lane = (row, proposal-of-half).
    float acc = b3;
#pragma unroll
    for (int h = 0; h < 8; ++h)
      acc = fmaf(lds[row * LROW + 8 * hi + h], w3r[h], acc);

    int aOut = aBase + row;
    if (aOut < numAll)
      out[(size_t)(hi ? p1 : p0) * numAll + aOut] = acc;
  }
}

// ---------------------------------------------------------------------------
extern "C" void kernel_launch(void* const* d_in, const int* in_sizes, int n_in,
                              void* d_out, int out_size, void* d_ws, size_t ws_size,
                              hipStream_t stream) {
  const float* prop  = (const float*)d_in[0];   // (128, 32)
  const float* feats = (const float*)d_in[1];   // (100000, 35)
  const float* Wdyn  = (const float*)d_in[2];   // (32, 369)
  const float* bdyn  = (const float*)d_in[3];   // (369,)
  float* outp   = (float*)d_out;                // (128, 100000)
  float* params = (float*)d_ws;                 // (128, 369) scratch

  const int numProp = in_sizes[0] / IN_C;       // 128
  const int numAll  = in_sizes[1] / CH;         // 100000

  const int totalP = numProp * PCOUNT;
  dynconv_params<<<(totalP + 255) / 256, 256, 0, stream>>>(
      prop, Wdyn, bdyn, params, numProp);

  const int numTiles = (numAll + 15) / 16;      // 6250
  const int pairs    = numProp / 2;             // 64
  int ygrid = (numTiles + 7) / 8;
  if (ygrid > 64) ygrid = 64;                   // 64x64 = 4096 blocks, 8 waves each
  dynconv_mlp<<<dim3(pairs, ygrid), 256, 0, stream>>>(
      feats, params, outp, numAll, numTiles);
}